// Neuron_75222057222206
// MI455X (gfx1250) — compile-verified
//
#include <hip/hip_runtime.h>
#include <hip/hip_bf16.h>

// LIF spiking neuron forward scan, MI455X (gfx1250).
// x: [B=32, T=32, N=65536] fp32, tau: [1] fp32, out: [B, T, N] fp32 spikes.
// Bandwidth-bound (512 MB total traffic, ~22 us floor at 23.3 TB/s).

typedef float v4f __attribute__((ext_vector_type(4)));

constexpr int kB = 32;
constexpr int kT = 32;
constexpr int kN = 65536;
constexpr int kN4 = kN / 4;              // 16384 float4 per (b, t) row
constexpr int kThreads = 256;
constexpr int kCols = kB * kN4;          // 524288 float4 columns
constexpr int kPipe = 4;                 // async tiles in flight per wave
constexpr long kRowBytes = (long)kN4 * 16;  // 256 KB stride per time step

// ---- CDNA5 async global->LDS copy (ASYNCcnt path), via inline asm ---------
// VGLOBAL encoding: VDST = LDS byte address VGPR (low 32 bits of generic
// shared pointer), VADDR = 64-bit global base VGPR pair, signed 24-bit
// immediate IOFFSET carries the time-step stride (max 31*256KB < 8MB).
__device__ __forceinline__ void async_copy_b128(const void* gbase, void* lds_generic,
                                                long ioffset) {
  unsigned lds_off = (unsigned)(unsigned long long)lds_generic;
  asm volatile("global_load_async_to_lds_b128 %0, %1, off offset:%2"
               :
               : "v"(lds_off), "v"(gbase), "i"(ioffset)
               : "memory");
}

__device__ __forceinline__ void wait_asynccnt(int n) {
  switch (n) {
    case 0: asm volatile("s_wait_asynccnt 0x0" ::: "memory"); break;
    case 1: asm volatile("s_wait_asynccnt 0x1" ::: "memory"); break;
    case 2: asm volatile("s_wait_asynccnt 0x2" ::: "memory"); break;
    default: asm volatile("s_wait_asynccnt 0x3" ::: "memory"); break;
  }
}

// ---------------------------------------------------------------------------
__global__ __launch_bounds__(kThreads) void lif_fwd_kernel(
    const float* __restrict__ xf, const float* __restrict__ taup,
    float* __restrict__ of) {
  const int tid = threadIdx.x;
  const int c = blockIdx.x * kThreads + tid;   // column id in [0, kCols)
  const int b = c >> 14;                       // c / kN4   (kN4 = 2^14)
  const int j = c & (kN4 - 1);                 // c % kN4
  const long base4 = (long)b * ((long)kT * kN4) + j;

  const v4f* __restrict__ x4 = (const v4f*)xf;
  v4f* __restrict__ o4 = (v4f*)of;

  float tc = taup[0];
  tc = fminf(fmaxf(tc, 0.0f), 1.0f);           // clamp learned decay to [0,1]

  v4f u = {0.0f, 0.0f, 0.0f, 0.0f};

  // 4-deep LDS ring: 4 * 256 * 16 B = 16 KB / block (320 KB per WGP).
  // Each wave reads back only LDS it wrote itself -> no barriers, only ASYNCcnt.
  __shared__ v4f buf[kPipe][kThreads];

  const v4f* gbase = x4 + base4;               // one base address for all loads

  // Prologue: stage tiles 0 .. kPipe-2 (3 tiles in flight).
#pragma unroll
  for (int p = 0; p < kPipe - 1; ++p) {
    async_copy_b128(gbase, &buf[p][tid], (long)p * kRowBytes);
  }

#pragma unroll
  for (int t = 0; t < kT; ++t) {
    const int issue = t + kPipe - 1;
    if (issue < kT) {
      async_copy_b128(gbase, &buf[issue & (kPipe - 1)][tid], (long)issue * kRowBytes);
    }
    // Outstanding tiles: t .. min(issue, kT-1); in-order decrement means
    // waiting for (outstanding-1) guarantees tile t is resident in LDS.
    const int last = (issue < kT) ? issue : (kT - 1);
    wait_asynccnt(last - t);                   // 3 steady-state; 2,1,0 at tail

    const v4f xv = buf[t & (kPipe - 1)][tid];

    v4f o;
#pragma unroll
    for (int i = 0; i < 4; ++i) {
      const float uu = fmaf(tc, u[i], xv[i]);  // leaky integrate
      const bool s = uu > 1.0f;                // threshold spike (THRESHOLD=1)
      o[i] = s ? 1.0f : 0.0f;
      u[i] = s ? 0.0f : uu;                    // hard reset to zero
    }

    // Write-once streaming output: non-temporal to keep L2 clean.
    __builtin_nontemporal_store(o, o4 + base4 + (long)t * kN4);
  }
}

// ---------------------------------------------------------------------------
extern "C" void kernel_launch(void* const* d_in, const int* in_sizes, int n_in,
                              void* d_out, int out_size, void* d_ws, size_t ws_size,
                              hipStream_t stream) {
  (void)in_sizes; (void)n_in; (void)out_size; (void)d_ws; (void)ws_size;
  const float* x = (const float*)d_in[0];
  const float* tau = (const float*)d_in[1];
  float* out = (float*)d_out;

  dim3 block(kThreads);
  dim3 grid(kCols / kThreads);   // 2048 blocks x 256 threads, wave32
  lif_fwd_kernel<<<grid, block, 0, stream>>>(x, tau, out);
}